// GumbelSinkhornNetwork_56891136803590
// MI455X (gfx1250) — compile-verified
//
#include <hip/hip_runtime.h>
#include <cstdint>

// ---------------------------------------------------------------------------
// Gumbel-Sinkhorn, MI455X (gfx1250).
//
// la0 = (log_alpha - log(EPS - log(noise+EPS))) / TEMP   (written once to d_out)
// 20x { R[i] += log sum_j exp(la0 - R[i] - C[j]);        (row normalize)
//       C[j] += log sum_i exp(la0 - R_new[i] - C[j]) }   (col normalize)
// out = exp(la0 - R[i] - C[j])
//
// The 512MB matrix is read-only after init; only the [B,N] vectors R,C change.
// Row+col passes are fused: a wave holds one full row (32 f32/lane) in VGPRs,
// computes the row LSE; the row-normalized probs e/s are the column-sum
// contributions for the C update (scaled by rcp(s), no extra exps).
// Per-block column sums: rows staged in LDS, then reduced with ones-matrix
// V_WMMA_F32_16X16X4_F32 (D = ones x B + C => D rows = colsum(B)), then one
// global_atomic_add_f32 per column per block. A tiny kernel finishes
// C[j] += log(acc[j]).
//
// CDNA5-specific: TDM tensor_load_to_lds stages the per-batch C vector (4KB)
// into LDS (TENSORcnt + s_wait_tensorcnt), WMMA column reduction, wave32
// shuffles, NT hints on the 512MB stream.
//
// d_ws layout: R [0, B*N) | C [B*N, 2B*N) | Cacc [2B*N, 3B*N)  == 1.5 MB.
// ---------------------------------------------------------------------------

#define BB 128
#define NN 1024
#define N_ITERS 20

typedef float f4 __attribute__((ext_vector_type(4)));
typedef float v2f __attribute__((ext_vector_type(2)));
typedef float v8f __attribute__((ext_vector_type(8)));
typedef unsigned int tdm_u32x4 __attribute__((ext_vector_type(4)));
typedef int tdm_i32x8 __attribute__((ext_vector_type(8)));
typedef int tdm_i32x4 __attribute__((ext_vector_type(4)));

__device__ __forceinline__ void tdm_load_4kb_row(uint32_t lds_off, const float* src) {
  // D# group0: count=1 descriptor, lds addr, 57-bit global addr, type=2.
  uint64_t ga = (uint64_t)(uintptr_t)src;
  tdm_u32x4 g0;
  g0[0] = 1u;                                   // count=1, user mode
  g0[1] = lds_off;                              // lds_addr (bytes)
  g0[2] = (uint32_t)ga;                         // global_addr[31:0]
  g0[3] = (uint32_t)((ga >> 32) & 0x1FFFFFFu)   // global_addr[56:32]
        | (2u << 30);                           // type=2 ("image")
  // D# group1: data_size=4B, tensor 1024x1, tile 1024x1, stride0=1024.
  tdm_i32x8 g1;
  g1[0] = (int)(2u << 16);        // workgroup_mask=0 | data_size=2 (4 bytes)
  g1[1] = (int)(1024u << 16);     // atomic_barrier_addr=0 | tensor_dim0[15:0]=1024
  g1[2] = (int)(1u << 16);        // tensor_dim0[31:16]=0 | tensor_dim1[15:0]=1
  g1[3] = (int)(1024u << 16);     // tensor_dim1[31:16]=0 | tile_dim0=1024
  g1[4] = 1;                      // tile_dim1=1 | tile_dim2=0
  g1[5] = 1024;                   // tensor_dim0_stride[31:0]
  g1[6] = (int)(1024u << 16);     // stride0[47:32]=0 | tensor_dim1_stride[15:0]
  g1[7] = 0;
  tdm_i32x4 z = {0, 0, 0, 0};     // groups 2/3 unused (<=2D tensor)
#if defined(__clang_major__) && (__clang_major__ >= 23)
  tdm_i32x8 z8 = {0, 0, 0, 0, 0, 0, 0, 0};
  __builtin_amdgcn_tensor_load_to_lds(g0, g1, z, z, z8, 0);
#else
  __builtin_amdgcn_tensor_load_to_lds(g0, g1, z, z, 0);
#endif
}

// --- init: la0 = (log_alpha + gumbel)/TEMP, streamed float4, NT both ways ---
__global__ __launch_bounds__(256) void k_init(const float* __restrict__ A,
                                              const float* __restrict__ U,
                                              float* __restrict__ O) {
  const float EPSc = 1e-20f;
  size_t i = (size_t)blockIdx.x * 256 + threadIdx.x;   // over 33554432 float4s
  f4 a = __builtin_nontemporal_load(&((const f4*)A)[i]);
  f4 u = __builtin_nontemporal_load(&((const f4*)U)[i]);
  f4 o;
  o.x = (a.x - __logf(EPSc - __logf(u.x + EPSc))) * 10.0f;  // /TEMP == *10
  o.y = (a.y - __logf(EPSc - __logf(u.y + EPSc))) * 10.0f;
  o.z = (a.z - __logf(EPSc - __logf(u.z + EPSc))) * 10.0f;
  o.w = (a.w - __logf(EPSc - __logf(u.w + EPSc))) * 10.0f;
  __builtin_nontemporal_store(o, &((f4*)O)[i]);
}

__global__ __launch_bounds__(256) void k_zero(float* __restrict__ p) {
  p[(size_t)blockIdx.x * 256 + threadIdx.x] = 0.0f;
}

// --- fused Sinkhorn sweep: one wave per row; 8 rows per block --------------
__global__ __launch_bounds__(256) void k_sink_row(const float* __restrict__ la0,
                                                  float* __restrict__ R,
                                                  const float* __restrict__ C,
                                                  float* __restrict__ Cacc,
                                                  int first) {
  __shared__ float cs[NN];        // per-batch C vector (TDM-staged)
  __shared__ float prow[8 * NN];  // row-normalized P rows for column reduction
  const int t = threadIdx.x;
  const int wave = t >> 5, lane = t & 31;
  const int b = blockIdx.x >> 7;              // 128 row-blocks per batch
  const int row = (blockIdx.x & 127) * 8 + wave;
  const size_t gRow = (size_t)b * NN + row;

  if (!first && t < 32) {   // wave 0 issues the TDM DMA (EXEC-independent)
    tdm_load_4kb_row((uint32_t)(uintptr_t)cs, C + (size_t)b * NN);
    __builtin_amdgcn_s_wait_tensorcnt(0);
  }
  __syncthreads();

  const f4* rp = (const f4*)(la0 + gRow * NN);
  const f4* cp = (const f4*)cs;
  f4 v[8];
#pragma unroll
  for (int k = 0; k < 8; ++k)
    v[k] = __builtin_nontemporal_load(&rp[k * 32 + lane]);   // 512B/step coalesced

  float r0 = 0.0f, m = 0.0f;
  if (first) {
    // Iteration 0: raw exponents unbounded (|la0| up to ~460) -> max-stabilize.
    m = -3.4e38f;
#pragma unroll
    for (int k = 0; k < 8; ++k)
      m = fmaxf(m, fmaxf(fmaxf(v[k].x, v[k].y), fmaxf(v[k].z, v[k].w)));
#pragma unroll
    for (int off = 16; off; off >>= 1) m = fmaxf(m, __shfl_xor(m, off, 32));
  } else {
    // After a col-normalize, la0 - R_old - C_old <= 0: no max needed.
    r0 = R[gRow];
#pragma unroll
    for (int k = 0; k < 8; ++k) v[k] -= cp[k * 32 + lane];
  }

  const float sh = r0 + m;
  float s = 0.0f;
#pragma unroll
  for (int k = 0; k < 8; ++k) {
    f4 e;
    e.x = __expf(v[k].x - sh);
    e.y = __expf(v[k].y - sh);
    e.z = __expf(v[k].z - sh);
    e.w = __expf(v[k].w - sh);
    v[k] = e;
    s += (e.x + e.y) + (e.z + e.w);
  }
#pragma unroll
  for (int off = 16; off; off >>= 1) s += __shfl_xor(s, off, 32);

  if (lane == 0) R[gRow] = sh + __logf(s);          // row LSE update
  const float inv = __builtin_amdgcn_rcpf(s);       // e/s = row-normalized P

  // Stage this wave's normalized row into LDS (conflict-free b128 stores).
  f4* pr = (f4*)&prow[wave * NN];
#pragma unroll
  for (int k = 0; k < 8; ++k) {
    f4 p = v[k];
    p.x *= inv; p.y *= inv; p.z *= inv; p.w *= inv;
    pr[k * 32 + lane] = p;
  }
  __syncthreads();

  // Column reduction of the 8 x 1024 P block via ones-matrix WMMA:
  // D = ones(16x4) x B(4x16) + C  =>  every row of D = column sums of B.
  // B layout (f32, 4x16): lanes 0-15 hold K=0,1 in V0,V1; lanes 16-31 K=2,3.
  float* gacc = Cacc + (size_t)b * NN;
  const int half = lane >> 4;         // 0: lanes 0-15, 1: lanes 16-31
  const int cl = lane & 15;           // column within 16-wide chunk
  v2f ones = {1.0f, 1.0f};
#pragma unroll
  for (int q = 0; q < 8; ++q) {       // 8 chunks of 16 columns per wave
    const int c0 = (wave * 8 + q) * 16;
    const int cc = c0 + cl;
    v8f acc = {0.0f, 0.0f, 0.0f, 0.0f, 0.0f, 0.0f, 0.0f, 0.0f};
#pragma unroll
    for (int g = 0; g < 2; ++g) {     // rows 0-3 then rows 4-7
      const int r0g = g * 4 + half * 2;
      v2f bmat;
      bmat[0] = prow[(r0g + 0) * NN + cc];
      bmat[1] = prow[(r0g + 1) * NN + cc];
      acc = __builtin_amdgcn_wmma_f32_16x16x4_f32(false, ones, false, bmat,
                                                  (short)0, acc, false, false);
    }
    // D VGPR0: lanes 0-15 = M=0 row, lanes 16-31 = M=8 row (identical colsums).
    if (lane < 16) atomicAdd(&gacc[cc], acc[0]);    // global_atomic_add_f32
  }
}

// --- finish col normalize: C += log(acc); reset acc ------------------------
__global__ __launch_bounds__(256) void k_col_fix(float* __restrict__ C,
                                                 float* __restrict__ Cacc,
                                                 int first) {
  size_t j = (size_t)blockIdx.x * 256 + threadIdx.x;  // over B*N
  float a = Cacc[j];
  Cacc[j] = 0.0f;
  float c0 = first ? 0.0f : C[j];
  C[j] = c0 + __logf(a);
}

// --- out = exp(la0 - R[i] - C[j]) in place ---------------------------------
__global__ __launch_bounds__(256) void k_final(float* __restrict__ la0,
                                               const float* __restrict__ R,
                                               const float* __restrict__ C) {
  const int b = blockIdx.x >> 10;                  // blockIdx.x == b*N + i
  const float r = R[blockIdx.x];
  const f4* c4 = (const f4*)(C + (size_t)b * NN);
  f4* rowp = (f4*)(la0 + (size_t)blockIdx.x * NN);
  f4 v = __builtin_nontemporal_load(&rowp[threadIdx.x]);
  f4 c = c4[threadIdx.x];
  f4 o;
  o.x = __expf(v.x - r - c.x);
  o.y = __expf(v.y - r - c.y);
  o.z = __expf(v.z - r - c.z);
  o.w = __expf(v.w - r - c.w);
  __builtin_nontemporal_store(o, &rowp[threadIdx.x]);
}

extern "C" void kernel_launch(void* const* d_in, const int* in_sizes, int n_in,
                              void* d_out, int out_size, void* d_ws, size_t ws_size,
                              hipStream_t stream) {
  (void)in_sizes; (void)n_in; (void)out_size; (void)ws_size;
  const float* la = (const float*)d_in[0];   // log_alpha  [128,1024,1024] f32
  const float* nz = (const float*)d_in[1];   // noise      [128,1024,1024] f32
  float* la0 = (float*)d_out;                // 512MB: la0 scratch, then output
  float* R   = (float*)d_ws;                 // [B,N]
  float* C   = R + (size_t)BB * NN;          // [B,N]
  float* Acc = C + (size_t)BB * NN;          // [B,N] column-sum accumulator

  k_init<<<dim3(131072), dim3(256), 0, stream>>>(la, nz, la0);
  k_zero<<<dim3((BB * NN) / 256), dim3(256), 0, stream>>>(Acc);
  for (int it = 0; it < N_ITERS; ++it) {
    k_sink_row<<<dim3(BB * 128), dim3(256), 0, stream>>>(la0, R, C, Acc, it == 0);
    k_col_fix<<<dim3((BB * NN) / 256), dim3(256), 0, stream>>>(C, Acc, it == 0);
  }
  k_final<<<dim3(BB * NN), dim3(256), 0, stream>>>(la0, R, C);
}